// VectorQuantizer_17291538334229
// MI455X (gfx1250) — compile-verified
//
#include <hip/hip_runtime.h>

typedef float v2f __attribute__((ext_vector_type(2)));
typedef float v8f __attribute__((ext_vector_type(8)));

#define K_CODES    8192
#define C_DIM      256
#define N_TOT      16384
#define KB         32          // codebook rows staged in LDS per iteration
#define LDS_STRIDE 260         // 256 + 4 pad -> row stride == 4 mod 64 banks (conflict-free b64 reads)
#define WAVES      4
#define BLOCK      128
#define Q_ELEMS    4194304     // 16*256*32*32

// ---------------------------------------------------------------------------
// cnorm[k] = sum_c cb[k][c]^2
// ---------------------------------------------------------------------------
__global__ void vq_cnorm_kernel(const float* __restrict__ cb, float* __restrict__ cnorm) {
    int k = blockIdx.x * blockDim.x + threadIdx.x;
    const float4* row = (const float4*)(cb + k * C_DIM);
    float s = 0.f;
#pragma unroll 8
    for (int i = 0; i < C_DIM / 4; ++i) {
        float4 v = row[i];
        s = fmaf(v.x, v.x, s); s = fmaf(v.y, v.y, s);
        s = fmaf(v.z, v.z, s); s = fmaf(v.w, v.w, s);
    }
    cnorm[k] = s;
}

// ---------------------------------------------------------------------------
// Main: per-wave 16-row tile, fp32 WMMA GEMM vs codebook, running argmin,
// gather/scatter of quantized output, deterministic loss partials.
// ---------------------------------------------------------------------------
__global__ __launch_bounds__(BLOCK, 1)
void vq_main_kernel(const float* __restrict__ x, const float* __restrict__ cb,
                    const float* __restrict__ cnorm,
                    float* __restrict__ outq, float* __restrict__ outidx,
                    float* __restrict__ partial) {
    __shared__ float sB[KB * LDS_STRIDE];
    __shared__ float sCN[KB];
    __shared__ int   sIdx[WAVES][16];
    __shared__ float sLoss[WAVES];

    const int tid  = threadIdx.x;
    const int w    = tid >> 5;
    const int lane = tid & 31;
    const int m    = lane & 15;   // M row within tile / N col within tile
    const int kh   = lane >> 4;   // K-pair selector for A/B fragments

    const int n0   = (blockIdx.x * WAVES + w) * 16; // first flat-row of this wave's tile
    const int r    = n0 + m;                        // this lane's flat row
    const int bidx = r >> 10;                       // batch index (tile never crosses batch: 16 | 1024)
    const int hw   = r & 1023;
    const int base = bidx * (C_DIM * 1024) + hw;    // &x[b][0][hw] in BCHW

    // ---- Preload A fragments: flat[r][c], WMMA f32 16x4 layout.
    // lane(l): M = l%16 ; VGPR0: K = 2*(l/16) ; VGPR1: K = 2*(l/16)+1
    v2f A[64];
#pragma unroll
    for (int s = 0; s < 64; ++s) {
        int c = 4 * s + 2 * kh;
        A[s].x = x[base + c * 1024];
        A[s].y = x[base + (c + 1) * 1024];
    }

    float minv[8];
    int   mini[8];
#pragma unroll
    for (int j = 0; j < 8; ++j) { minv[j] = __builtin_inff(); mini[j] = 0; }

    // ---- Stream the codebook through LDS in 32-row blocks
    for (int k0 = 0; k0 < K_CODES; k0 += KB) {
        __syncthreads();
        for (int i = tid; i < KB * (C_DIM / 4); i += BLOCK) {
            int row = i >> 6;
            int f4  = i & 63;
            float4 v = ((const float4*)(cb + (k0 + row) * C_DIM))[f4];
            *(float4*)(sB + row * LDS_STRIDE + f4 * 4) = v;
        }
        if (tid < KB) sCN[tid] = cnorm[k0 + tid];
        __syncthreads();

#pragma unroll
        for (int t = 0; t < KB / 16; ++t) {
            // B fragment base: B[k][n] -> lane n = l%16 picks codebook row,
            // VGPR0: c = 4s + 2*(l/16), VGPR1: c+1  (mirrors A layout)
            const v2f* bp = (const v2f*)(sB + (16 * t + m) * LDS_STRIDE + 2 * kh);

            v8f acc = {0.f, 0.f, 0.f, 0.f, 0.f, 0.f, 0.f, 0.f};
#pragma unroll
            for (int s = 0; s < 64; ++s) {
                acc = __builtin_amdgcn_wmma_f32_16x16x4_f32(
                    false, A[s], false, bp[2 * s], (short)0, acc, false, false);
            }

            const float cn   = sCN[16 * t + m];
            const int   kcol = k0 + 16 * t + m;
#pragma unroll
            for (int j = 0; j < 8; ++j) {
                float cand = fmaf(-2.f, acc[j], cn);
                if (cand < minv[j]) { minv[j] = cand; mini[j] = kcol; }
            }
        }
    }

    // ---- Cross-lane min+argmin within each 16-lane half (C layout: VGPR j -> row j / j+8)
#pragma unroll
    for (int j = 0; j < 8; ++j) {
#pragma unroll
        for (int mask = 1; mask < 16; mask <<= 1) {
            float ov = __shfl_xor(minv[j], mask, 32);
            int   oi = __shfl_xor(mini[j], mask, 32);
            if (ov < minv[j] || (ov == minv[j] && oi < mini[j])) { minv[j] = ov; mini[j] = oi; }
        }
    }
    if (lane == 0) {
#pragma unroll
        for (int j = 0; j < 8; ++j) { sIdx[w][j] = mini[j]; outidx[n0 + j] = (float)mini[j]; }
    }
    if (lane == 16) {
#pragma unroll
        for (int j = 0; j < 8; ++j) { sIdx[w][8 + j] = mini[j]; outidx[n0 + 8 + j] = (float)mini[j]; }
    }
    __syncthreads();

    // ---- Gather codebook rows, scatter quantized BCHW, accumulate (q-x)^2.
    // lane pair (m, m+16) covers row n0+m: even/odd channels -> contiguous 64B half-wave stores.
    const int code   = sIdx[w][m];
    const int cbbase = code * C_DIM;
    float lsum = 0.f;
#pragma unroll 4
    for (int t = 0; t < 128; ++t) {
        int   cc   = 2 * t + kh;
        float qv   = cb[cbbase + cc];
        int   addr = base + cc * 1024;
        float xv   = x[addr];
        outq[addr] = qv;
        float d = qv - xv;
        lsum = fmaf(d, d, lsum);
    }
#pragma unroll
    for (int mask = 16; mask >= 1; mask >>= 1) lsum += __shfl_xor(lsum, mask, 32);
    if (lane == 0) sLoss[w] = lsum;
    __syncthreads();
    if (tid == 0) {
        float s = 0.f;
#pragma unroll
        for (int i = 0; i < WAVES; ++i) s += sLoss[i];
        partial[blockIdx.x] = s;
    }
}

// ---------------------------------------------------------------------------
// Deterministic final loss reduction over 256 block partials
// ---------------------------------------------------------------------------
__global__ void vq_finalize_kernel(const float* __restrict__ partial, float* __restrict__ outloss) {
    __shared__ float sm[256];
    sm[threadIdx.x] = partial[threadIdx.x];
    __syncthreads();
    for (int s = 128; s > 0; s >>= 1) {
        if (threadIdx.x < s) sm[threadIdx.x] += sm[threadIdx.x + s];
        __syncthreads();
    }
    if (threadIdx.x == 0)
        outloss[0] = 1.25f * sm[0] / (float)Q_ELEMS;  // commitment(0.25) + embedding(1.0)
}

extern "C" void kernel_launch(void* const* d_in, const int* in_sizes, int n_in,
                              void* d_out, int out_size, void* d_ws, size_t ws_size,
                              hipStream_t stream) {
    const float* x  = (const float*)d_in[0];   // (16,256,32,32) fp32
    const float* cb = (const float*)d_in[1];   // (8192,256) fp32
    float* out = (float*)d_out;                // [q(4194304) | loss(1) | idx(16384)]
    float* ws  = (float*)d_ws;
    float* cnorm   = ws;            // 8192 floats
    float* partial = ws + K_CODES;  // 256 floats

    vq_cnorm_kernel<<<K_CODES / 256, 256, 0, stream>>>(cb, cnorm);
    vq_main_kernel<<<N_TOT / (16 * WAVES), BLOCK, 0, stream>>>(
        x, cb, cnorm, out, out + Q_ELEMS + 1, partial);
    vq_finalize_kernel<<<1, 256, 0, stream>>>(partial, out + Q_ELEMS);
}